// CalibAttentionLayer_42296837931443
// MI455X (gfx1250) — compile-verified
//
#include <hip/hip_runtime.h>
#include <math.h>

#define IN_CH 512
#define HEADS 8
#define BPAD 16            // B columns padded 8 -> 16 (zeros) so EXEC stays all-ones
#define NEG_SLOPE 0.2f

typedef __attribute__((ext_vector_type(2))) float v2f;
typedef __attribute__((ext_vector_type(8))) float v8f;

// ---------------------------------------------------------------------------
// Monotonic float <-> uint mapping so we can use native u32 atomic max.
// ---------------------------------------------------------------------------
__device__ __forceinline__ unsigned encF(float f) {
  unsigned u = __float_as_uint(f);
  return (u & 0x80000000u) ? ~u : (u | 0x80000000u);
}
__device__ __forceinline__ float decF(unsigned u) {
  u = (u & 0x80000000u) ? (u & 0x7FFFFFFFu) : ~u;
  return __uint_as_float(u);
}

__device__ __forceinline__ float leaky(float v) {
  return v >= 0.f ? v : NEG_SLOPE * v;
}

// ---------------------------------------------------------------------------
// Kernel 1: h = x @ W via V_WMMA_F32_16X16X4_F32, plus per-head attention
// scalars a_src / a_dst.  One wave handles a 16-row tile of x; K=512 is
// consumed 4 at a time (128 WMMA issues per tile).  W staged in LDS padded
// to 16 columns with zeros -> no EXEC masking anywhere in the hot loop.
// ---------------------------------------------------------------------------
__global__ __launch_bounds__(256) void gat_gemm_kernel(
    const float* __restrict__ x, const float* __restrict__ W,
    const float* __restrict__ att_src, const float* __restrict__ att_dst,
    float* __restrict__ h, float* __restrict__ asrc, float* __restrict__ adst,
    int n)
{
  __shared__ float Wlds[IN_CH * BPAD];  // 32 KB, [k][col] with col 8..15 = 0
  for (int i = threadIdx.x; i < IN_CH * BPAD; i += blockDim.x) {
    const int k = i >> 4;
    const int c = i & 15;
    Wlds[i] = (c < HEADS) ? W[k * HEADS + c] : 0.f;
  }
  __syncthreads();

  const int lane = threadIdx.x & 31;
  const int wave = threadIdx.x >> 5;
  const int half = lane >> 4;   // 0: K=k,k+1   1: K=k+2,k+3
  const int l    = lane & 15;   // A: row-in-tile   B/C: column

  const int tile = blockIdx.x * (blockDim.x >> 5) + wave;
  const int row0 = tile * 16;
  if (row0 >= n) return;

  int arow = row0 + l;
  if (arow > n - 1) arow = n - 1;           // clamp (keeps EXEC all-ones)
  const float* __restrict__ xrow = x + (size_t)arow * IN_CH;
  const float* __restrict__ wcol = Wlds + (size_t)half * 2 * BPAD + l;

  v8f c = {0.f, 0.f, 0.f, 0.f, 0.f, 0.f, 0.f, 0.f};

  #pragma unroll 4
  for (int k = 0; k < IN_CH; k += 4) {
    const int ka = k + half * 2;
    v2f a;
    a.x = xrow[ka];
    a.y = xrow[ka + 1];
    v2f b;
    b.x = wcol[k * BPAD];            // W[ka][l]   (0 for cols 8..15)
    b.y = wcol[k * BPAD + BPAD];     // W[ka+1][l]
    // D = A(16x4,f32) * B(4x16,f32) + C(16x16,f32)
    c = __builtin_amdgcn_wmma_f32_16x16x4_f32(
        /*neg_a=*/false, a, /*neg_b=*/false, b,
        /*c_mod=*/(short)0, c, /*reuse_a=*/false, /*reuse_b=*/false);
  }

  // C/D layout: VGPR i -> lanes 0-15: M=i, N=lane; lanes 16-31: M=i+8, N=lane-16
  if (l < HEADS) {
    const float as = att_src[l];
    const float ad = att_dst[l];
    #pragma unroll
    for (int i = 0; i < 8; ++i) {
      const int r = row0 + i + half * 8;
      if (r < n) {
        const float hv = c[i];
        const size_t o = (size_t)r * HEADS + l;
        h[o]    = hv;
        asrc[o] = hv * as;
        adst[o] = hv * ad;
      }
    }
  }
}

// ---------------------------------------------------------------------------
// Kernel 2: initialize accumulators (emax=-inf encoded, denom=0, numer=0)
// ---------------------------------------------------------------------------
__global__ void gat_init_kernel(unsigned* __restrict__ emaxEnc,
                                float* __restrict__ denom,
                                float* __restrict__ numer, size_t cnt)
{
  size_t i = (size_t)blockIdx.x * blockDim.x + threadIdx.x;
  if (i < cnt) {
    emaxEnc[i] = 0x007FFFFFu;   // encF(-inf)
    denom[i]   = 0.f;
    numer[i]   = 0.f;
  }
}

// ---------------------------------------------------------------------------
// Kernel 3: per-edge segment max of leaky_relu(a_src[src] + a_dst[dst])
// Self-loops appended as edge ids [nE, nE+n).
// ---------------------------------------------------------------------------
__global__ __launch_bounds__(256) void gat_edge_max_kernel(
    const int* __restrict__ ei,
    const float* __restrict__ asrc, const float* __restrict__ adst,
    unsigned* __restrict__ emaxEnc, int nE, int n)
{
  const int e = blockIdx.x * blockDim.x + threadIdx.x;
  const int etot = nE + n;
  if (e >= etot) return;
  int src, dst;
  if (e < nE) { src = ei[e]; dst = ei[nE + e]; }
  else        { src = dst = e - nE; }

  const float4* ps = (const float4*)(asrc + (size_t)src * HEADS);
  const float4* pd = (const float4*)(adst + (size_t)dst * HEADS);
  const float4 s0 = ps[0], s1 = ps[1];
  const float4 d0 = pd[0], d1 = pd[1];
  float v[HEADS] = { s0.x + d0.x, s0.y + d0.y, s0.z + d0.z, s0.w + d0.w,
                     s1.x + d1.x, s1.y + d1.y, s1.z + d1.z, s1.w + d1.w };
  unsigned* base = emaxEnc + (size_t)dst * HEADS;
  #pragma unroll
  for (int hh = 0; hh < HEADS; ++hh)
    atomicMax(base + hh, encF(leaky(v[hh])));
}

// ---------------------------------------------------------------------------
// Kernel 4: fused softmax-exp + weighted-message accumulation.
//   denom[dst,h] += exp(e - emax);  numer[dst,h] += exp(e - emax) * h[src,h]
// ---------------------------------------------------------------------------
__global__ __launch_bounds__(256) void gat_edge_acc_kernel(
    const int* __restrict__ ei,
    const float* __restrict__ asrc, const float* __restrict__ adst,
    const float* __restrict__ hmat, const unsigned* __restrict__ emaxEnc,
    float* __restrict__ denom, float* __restrict__ numer, int nE, int n)
{
  const int e = blockIdx.x * blockDim.x + threadIdx.x;
  const int etot = nE + n;
  if (e >= etot) return;
  int src, dst;
  if (e < nE) { src = ei[e]; dst = ei[nE + e]; }
  else        { src = dst = e - nE; }

  const float4* ps = (const float4*)(asrc + (size_t)src * HEADS);
  const float4* pd = (const float4*)(adst + (size_t)dst * HEADS);
  const float4* ph = (const float4*)(hmat + (size_t)src * HEADS);
  const float4 s0 = ps[0], s1 = ps[1];
  const float4 d0 = pd[0], d1 = pd[1];
  const float4 h0 = ph[0], h1 = ph[1];
  float v[HEADS]  = { s0.x + d0.x, s0.y + d0.y, s0.z + d0.z, s0.w + d0.w,
                      s1.x + d1.x, s1.y + d1.y, s1.z + d1.z, s1.w + d1.w };
  float hv[HEADS] = { h0.x, h0.y, h0.z, h0.w, h1.x, h1.y, h1.z, h1.w };

  const unsigned* me = emaxEnc + (size_t)dst * HEADS;
  float* dden = denom + (size_t)dst * HEADS;
  float* dnum = numer + (size_t)dst * HEADS;
  #pragma unroll
  for (int hh = 0; hh < HEADS; ++hh) {
    const float ee = expf(leaky(v[hh]) - decF(me[hh]));
    atomicAdd(dden + hh, ee);
    atomicAdd(dnum + hh, ee * hv[hh]);
  }
}

// ---------------------------------------------------------------------------
// Kernel 5: per-node finalize: softmax divide + gat_bias + calibration head.
// ---------------------------------------------------------------------------
__global__ __launch_bounds__(256) void gat_finalize_kernel(
    const float* __restrict__ numer, const float* __restrict__ denom,
    const float* __restrict__ gat_bias, const float* __restrict__ lin_w,
    const float* __restrict__ lin_b, const float* __restrict__ bias,
    float* __restrict__ out, int n)
{
  const int i = blockIdx.x * blockDim.x + threadIdx.x;
  if (i >= n) return;
  const float* nu = numer + (size_t)i * HEADS;
  const float* de = denom + (size_t)i * HEADS;
  float z = lin_b[0];
  #pragma unroll
  for (int hh = 0; hh < HEADS; ++hh) {
    const float oh = nu[hh] / (de[hh] + 1e-16f) + gat_bias[hh];
    z += oh * lin_w[hh];
  }
  out[i] = fmaxf(z, 0.f) + bias[0];
}

// ---------------------------------------------------------------------------
// Host-side launcher
// ---------------------------------------------------------------------------
extern "C" void kernel_launch(void* const* d_in, const int* in_sizes, int n_in,
                              void* d_out, int out_size, void* d_ws, size_t ws_size,
                              hipStream_t stream) {
  const float* x        = (const float*)d_in[0];
  const int*   ei       = (const int*)d_in[1];
  const float* W        = (const float*)d_in[2];
  const float* att_src  = (const float*)d_in[3];
  const float* att_dst  = (const float*)d_in[4];
  const float* gat_bias = (const float*)d_in[5];
  const float* lin_w    = (const float*)d_in[6];
  const float* lin_b    = (const float*)d_in[7];
  const float* bias     = (const float*)d_in[8];
  float* out = (float*)d_out;

  const int n  = in_sizes[0] / IN_CH;
  const int nE = in_sizes[1] / 2;

  // workspace layout (each segment 256B aligned)
  const size_t nh  = (size_t)n * HEADS;
  const size_t seg = (nh * sizeof(float) + 255) & ~(size_t)255;
  char* ws = (char*)d_ws;
  float*    h_mat   = (float*)   (ws + 0 * seg);
  float*    asrc    = (float*)   (ws + 1 * seg);
  float*    adst    = (float*)   (ws + 2 * seg);
  unsigned* emaxEnc = (unsigned*)(ws + 3 * seg);
  float*    denom   = (float*)   (ws + 4 * seg);
  float*    numer   = (float*)   (ws + 5 * seg);

  // 1) GEMM + attention scalars (1 wave per 16-row tile, 8 waves per block)
  const int tiles  = (n + 15) / 16;
  const int blocks = (tiles + 7) / 8;
  gat_gemm_kernel<<<blocks, 256, 0, stream>>>(x, W, att_src, att_dst,
                                              h_mat, asrc, adst, n);

  // 2) init accumulators
  {
    const int b = (int)((nh + 255) / 256);
    gat_init_kernel<<<b, 256, 0, stream>>>(emaxEnc, denom, numer, nh);
  }

  // 3) segment max over edges (+ self loops)
  const int etot = nE + n;
  const int eb = (etot + 255) / 256;
  gat_edge_max_kernel<<<eb, 256, 0, stream>>>(ei, asrc, adst, emaxEnc, nE, n);

  // 4) fused exp + numerator/denominator accumulation
  gat_edge_acc_kernel<<<eb, 256, 0, stream>>>(ei, asrc, adst, h_mat, emaxEnc,
                                              denom, numer, nE, n);

  // 5) finalize
  gat_finalize_kernel<<<(n + 255) / 256, 256, 0, stream>>>(
      numer, denom, gat_bias, lin_w, lin_b, bias, out, n);
}